// NFM_8014408974952
// MI455X (gfx1250) — compile-verified
//
#include <hip/hip_runtime.h>

typedef __attribute__((ext_vector_type(2))) float v2f;
typedef __attribute__((ext_vector_type(8))) float v8f;

#define BATCH 16384
#define FIELDS 50
#define EMB 64
#define HID 200
#define ZP 256                 // padded channel stride for z (coalesced stats)
#define NT ((HID + 15) / 16)   // 13 N-tiles
#define MT (BATCH / 16)        // 1024 M-tiles
#define NSTAT_BLOCKS 128

// ---------------------------------------------------------------------------
// K1: embedding gather + FM bi-interaction. One wave per batch row; each lane
// holds 2 of the 64 channels (float2 -> one coalesced 256B line pair per row).
// Indices preloaded with two coalesced loads and broadcast with __shfl so the
// fully-unrolled loop can keep many random gathers in flight (HBM-bound).
// ---------------------------------------------------------------------------
__global__ void nfm_fm_gather(const int* __restrict__ x,
                              const float* __restrict__ emb,
                              float* __restrict__ bi) {
  const int lane = threadIdx.x & 31;
  const int row  = blockIdx.x * (blockDim.x >> 5) + (threadIdx.x >> 5);
  if (row >= BATCH) return;
  const int* xr = x + row * FIELDS;
  const int i0 = xr[lane];                                   // f = 0..31
  const int i1 = (lane < FIELDS - 32) ? xr[32 + lane] : 0;   // f = 32..49
  float sx = 0.f, sy = 0.f, qx = 0.f, qy = 0.f;
  #pragma unroll
  for (int f = 0; f < FIELDS; ++f) {
    const int idx = __shfl(f < 32 ? i0 : i1, f & 31, 32);
    const float2 e = ((const float2*)(emb + (size_t)idx * EMB))[lane];
    sx += e.x; sy += e.y;
    qx += e.x * e.x; qy += e.y * e.y;
  }
  float2 o;
  o.x = 0.5f * (sx * sx - qx);
  o.y = 0.5f * (sy * sy - qy);
  ((float2*)(bi + (size_t)row * EMB))[lane] = o;
}

// ---------------------------------------------------------------------------
// K2a: per-block partial batch stats, fully coalesced. CP = padded channel
// stride (64 for bi, 256 for padded z). 256 threads = CP channels x (256/CP)
// rows per iteration; register accumulation, LDS fold, deterministic.
// ---------------------------------------------------------------------------
template <int CP>
__global__ void nfm_bn_stats_partial(const float* __restrict__ src,
                                     int rowsPerBlock,
                                     float* __restrict__ part) {
  constexpr int RPI = 256 / CP;
  const int c  = threadIdx.x % CP;
  const int rl = threadIdx.x / CP;
  const int r0 = blockIdx.x * rowsPerBlock;
  float s = 0.f, q = 0.f;
  for (int r = r0 + rl; r < r0 + rowsPerBlock; r += RPI) {
    const float v = src[(size_t)r * CP + c];
    s += v; q += v * v;
  }
  if (RPI > 1) {
    __shared__ float ls[256], lq[256];
    ls[threadIdx.x] = s; lq[threadIdx.x] = q;
    __syncthreads();
    if (rl == 0) {
      #pragma unroll
      for (int j = 1; j < RPI; ++j) { s += ls[c + j * CP]; q += lq[c + j * CP]; }
    }
  }
  if (rl == 0) {
    part[(size_t)blockIdx.x * (2 * CP) + c]      = s;
    part[(size_t)blockIdx.x * (2 * CP) + CP + c] = q;
  }
}

// K2b: combine block partials -> mean, 1/sqrt(var+eps). One small block.
template <int CP>
__global__ void nfm_bn_stats_final(const float* __restrict__ part, int nblocks,
                                   int C, float* __restrict__ mean,
                                   float* __restrict__ inv) {
  const int c = threadIdx.x;
  if (c >= C) return;
  float s = 0.f, q = 0.f;
  for (int b = 0; b < nblocks; ++b) {
    s += part[(size_t)b * (2 * CP) + c];
    q += part[(size_t)b * (2 * CP) + CP + c];
  }
  const float m = s / (float)BATCH;
  mean[c] = m;
  inv[c]  = rsqrtf(q / (float)BATCH - m * m + 1e-5f);  // biased variance
}

// ---------------------------------------------------------------------------
// K3: BN1 + ReLU applied in place on bi (C=64 hardcoded -> cheap & mask).
// ---------------------------------------------------------------------------
__global__ void nfm_bn_relu_inplace(float* __restrict__ buf,
                                    const float* __restrict__ mean,
                                    const float* __restrict__ inv,
                                    const float* __restrict__ gamma,
                                    const float* __restrict__ beta) {
  const int i = blockIdx.x * blockDim.x + threadIdx.x;
  if (i >= BATCH * EMB) return;
  const int c = i & (EMB - 1);
  const float v = (buf[i] - mean[c]) * inv[c] * gamma[c] + beta[c];
  buf[i] = v > 0.f ? v : 0.f;
}

// ---------------------------------------------------------------------------
// K4: z = h0 @ w1 + b1 via V_WMMA_F32_16X16X4_F32 (exact fp32 matrix path).
// One wave per 16x16 output tile; 16 WMMAs over K=64. Out-of-range N columns
// are CLAMPED for loads (column n of D depends only on column n of B, so the
// garbage columns are simply not stored) -> no exec-mask churn in the loop,
// and A/B loads become immediate-offset clauses off fixed pointers.
// ---------------------------------------------------------------------------
__global__ void nfm_gemm_wmma(const float* __restrict__ h0,
                              const float* __restrict__ w1,
                              const float* __restrict__ b1,
                              float* __restrict__ z) {
  const int lane = threadIdx.x;                       // blockDim.x == 32
  const int tile = blockIdx.x * blockDim.y + threadIdx.y;
  const int mt = tile / NT;
  const int nt = tile % NT;
  const int row0 = mt * 16;
  const int n0   = nt * 16;
  const int m    = lane & 15;
  const int koff = (lane >> 4) * 2;                   // 0 or 2
  const int n    = n0 + (lane & 15);
  const int ncl  = n < HID ? n : HID - 1;             // clamp, discarded at store

  const float* ap = h0 + (size_t)(row0 + m) * EMB + koff;
  const float* bp = w1 + (size_t)koff * HID + ncl;

  v8f c = {};
  #pragma unroll
  for (int k0 = 0; k0 < EMB; k0 += 4) {
    v2f a, b;
    a.x = ap[k0];
    a.y = ap[k0 + 1];
    b.x = bp[k0 * HID];            // immediate offsets, stride 3200B
    b.y = bp[(k0 + 1) * HID];
    // 8 args: (neg_a, A, neg_b, B, c_mod, C, reuse_a, reuse_b)
    c = __builtin_amdgcn_wmma_f32_16x16x4_f32(false, a, false, b,
                                              (short)0, c, false, false);
  }

  if (n < HID) {
    const float bias = b1[n];
    const int mbase = row0 + ((lane >> 4) * 8);
    #pragma unroll
    for (int r = 0; r < 8; ++r)
      z[(size_t)(mbase + r) * ZP + n] = c[r] + bias;   // padded stride 256
  }
}

// ---------------------------------------------------------------------------
// K5: fused BN2 + ReLU + dot(w_last) + sigmoid. One wave per row; wave32
// shuffle-xor reduction.
// ---------------------------------------------------------------------------
__global__ void nfm_final(const float* __restrict__ z,
                          const float* __restrict__ mean2,
                          const float* __restrict__ inv2,
                          const float* __restrict__ g,
                          const float* __restrict__ be,
                          const float* __restrict__ wl,
                          const float* __restrict__ bl,
                          float* __restrict__ out) {
  const int lane = threadIdx.x & 31;
  const int row  = blockIdx.x * (blockDim.x >> 5) + (threadIdx.x >> 5);
  if (row >= BATCH) return;
  float acc = 0.f;
  #pragma unroll
  for (int k0 = 0; k0 < HID; k0 += 32) {
    const int k = k0 + lane;
    if (k < HID) {
      float v = (z[(size_t)row * ZP + k] - mean2[k]) * inv2[k] * g[k] + be[k];
      v = v > 0.f ? v : 0.f;
      acc += v * wl[k];
    }
  }
  #pragma unroll
  for (int off = 16; off > 0; off >>= 1)
    acc += __shfl_xor(acc, off, 32);
  if (lane == 0)
    out[row] = 1.f / (1.f + expf(-(acc + bl[0])));
}

// ---------------------------------------------------------------------------
extern "C" void kernel_launch(void* const* d_in, const int* in_sizes, int n_in,
                              void* d_out, int out_size, void* d_ws, size_t ws_size,
                              hipStream_t stream) {
  const int*   x      = (const int*)  d_in[0];
  const float* emb    = (const float*)d_in[1];
  const float* w1     = (const float*)d_in[2];
  const float* b1     = (const float*)d_in[3];
  const float* w_last = (const float*)d_in[4];
  const float* b_last = (const float*)d_in[5];
  const float* g_bi   = (const float*)d_in[6];
  const float* be_bi  = (const float*)d_in[7];
  const float* g_h    = (const float*)d_in[8];
  const float* be_h   = (const float*)d_in[9];
  float* out = (float*)d_out;

  char* ws = (char*)d_ws;
  float* bi   = (float*)ws;                                   // 16384*64  (4 MB), BN'd in place -> h0
  float* z    = (float*)(ws + (size_t)BATCH * EMB * 4);       // 16384*256 (16.8 MB, padded)
  float* part = (float*)(ws + (size_t)BATCH * EMB * 4
                            + (size_t)BATCH * ZP * 4);        // 128*512 partials (256 KB)
  float* stats = part + (size_t)NSTAT_BLOCKS * 2 * ZP;
  float* mean1 = stats;          // 64
  float* inv1  = mean1 + EMB;    // 64
  float* mean2 = inv1 + EMB;     // 200
  float* inv2  = mean2 + HID;    // 200

  // 1) gather + FM bi-interaction (HBM-bound hot spot)
  nfm_fm_gather<<<BATCH / 8, 256, 0, stream>>>(x, emb, bi);
  // 2) BN1 batch stats (coalesced two-phase reduction)
  nfm_bn_stats_partial<EMB><<<NSTAT_BLOCKS, 256, 0, stream>>>(
      bi, BATCH / NSTAT_BLOCKS, part);
  nfm_bn_stats_final<EMB><<<1, EMB, 0, stream>>>(part, NSTAT_BLOCKS, EMB,
                                                 mean1, inv1);
  // 3) BN1 + ReLU in place -> h0
  nfm_bn_relu_inplace<<<(BATCH * EMB + 255) / 256, 256, 0, stream>>>(
      bi, mean1, inv1, g_bi, be_bi);
  // 4) z = h0 @ w1 + b1 (fp32 WMMA), padded stride 256
  nfm_gemm_wmma<<<(MT * NT + 3) / 4, dim3(32, 4), 0, stream>>>(bi, w1, b1, z);
  // 5) BN2 batch stats (coalesced over padded stride; pad channels discarded)
  nfm_bn_stats_partial<ZP><<<NSTAT_BLOCKS, 256, 0, stream>>>(
      z, BATCH / NSTAT_BLOCKS, part);
  nfm_bn_stats_final<ZP><<<1, ZP, 0, stream>>>(part, NSTAT_BLOCKS, HID,
                                               mean2, inv2);
  // 6) BN2 + ReLU + dot + sigmoid
  nfm_final<<<BATCH / 8, 256, 0, stream>>>(z, mean2, inv2, g_h, be_h,
                                           w_last, b_last, out);
}